// MVFeedforwardBlock_41274635715140
// MI455X (gfx1250) — compile-verified
//
#include <hip/hip_runtime.h>
#include <math.h>

#define BLOCK 256

typedef __attribute__((ext_vector_type(2))) float v2f;
typedef __attribute__((ext_vector_type(8))) float v8f;

// ---------------------------------------------------------------------------
// Compile-time Clifford algebra tables, metric (0,1,1,1), blade order matching
// the reference BLADES list.
// ---------------------------------------------------------------------------
namespace ga {
constexpr int MASK[16]  = {0,1,2,4,8,3,5,9,6,10,12,7,11,13,14,15};  // idx -> bitmask
constexpr int IDXOF[16] = {0,1,2,5,3,6,8,11,4,7,9,12,10,13,14,15};  // bitmask -> idx
constexpr int GRADE[16] = {0,1,1,1,1,2,2,2,2,2,2,3,3,3,3,4};
constexpr int PRE[16]   = {-1,0,-1,-1,-1,2,3,4,-1,-1,-1,8,9,10,-1,14}; // c with {0}|c == d
constexpr int NE[16]    = {1,0,1,1,1,0,0,0,1,1,1,0,0,0,1,0};           // blade w/o e0

constexpr int popc4(int x){ return (x&1)+((x>>1)&1)+((x>>2)&1)+((x>>3)&1); }
constexpr int reord(int a, int b){            // (-1)^(#transpositions to merge-sort a,b)
  int s = 0;
  for (int i = 0; i < 4; ++i) if ((b>>i)&1) s += popc4(a >> (i+1));
  return (s & 1) ? -1 : 1;
}
struct T16 { signed char k[16][16]; signed char s[16][16]; };
constexpr T16 mkGP(){
  T16 t{};
  for (int i=0;i<16;++i) for (int j=0;j<16;++j){
    int a = MASK[i], b = MASK[j];
    if (a & b & 1) { t.s[i][j]=0; t.k[i][j]=0; }          // e0*e0 = 0
    else { t.s[i][j]=(signed char)reord(a,b); t.k[i][j]=(signed char)IDXOF[a^b]; }
  }
  return t;
}
constexpr int compIdx(int i){ return IDXOF[15 ^ MASK[i]]; }
constexpr int dsgn(int i){ return reord(MASK[i], 15 ^ MASK[i]); } // sign of blade ^ complement
constexpr T16 mkJOIN(){   // JOIN[i,j,k] = DINV[k,c] WEDGE[a,b,c] D[a,i] D[b,j]
  T16 t{};
  for (int i=0;i<16;++i) for (int j=0;j<16;++j){
    int a = compIdx(i), b = compIdx(j);
    int ma = MASK[a], mb = MASK[b];
    if (ma & mb) { t.s[i][j]=0; t.k[i][j]=0; }
    else {
      int c = IDXOF[ma^mb];
      int k = compIdx(c);
      int s = dsgn(i)*dsgn(j)*reord(ma,mb)*dsgn(k);
      t.s[i][j]=(signed char)s; t.k[i][j]=(signed char)k;
    }
  }
  return t;
}
constexpr T16 GP = mkGP();
constexpr T16 JN = mkJOIN();
} // namespace ga

// Runtime-indexable packed tables (avoid scratch from dynamic register-array idx)
__device__ __forceinline__ int grade_of(int d){ return (int)((0x4333322222211110ULL >> (4*d)) & 0xFULL); }
__device__ __forceinline__ int pre_of(int d){ return (int)((0xEFA98FFF432FFF0FULL >> (4*d)) & 0xFULL); } // 0xF == none

// mv_linear matrix entry M[row=c][col=d] built from scalar weights via select chains
__device__ __forceinline__ float mval(int row, int col,
                                      float w0,float w1,float w2,float w3,float w4,
                                      float v1,float v2,float v3){
  int g = grade_of(col);
  float w = w0; w=(g==1)?w1:w; w=(g==2)?w2:w; w=(g==3)?w3:w; w=(g==4)?w4:w;
  float m = (row == col) ? w : 0.0f;
  int p = pre_of(col);
  float vv = (g==1)?v1:((g==2)?v2:v3);
  m += ((p != 15) && (p == row) && (g <= 3)) ? vv : 0.0f;
  return m;
}

// Per-token nonlinear core: x -> (gp, join) vectors (both length 16)
__device__ __forceinline__ void token_core(const float X[16],
                                           const float W1[5], const float V1[4], float B1,
                                           float GPV[16], float JNV[16]){
  float nrm2 = 1e-5f;
  #pragma unroll
  for (int d=0; d<16; ++d) if (ga::NE[d]) nrm2 += X[d]*X[d];
  float inv = 1.0f / sqrtf(nrm2);
  float XN[16];
  #pragma unroll
  for (int d=0; d<16; ++d) XN[d] = X[d]*inv;

  float XP[16];
  #pragma unroll
  for (int d=0; d<16; ++d){
    float t = W1[ga::GRADE[d]] * XN[d];
    if (ga::PRE[d] >= 0 && ga::GRADE[d] <= 3) t += V1[ga::GRADE[d]] * XN[ga::PRE[d]];
    XP[d] = t;
  }
  XP[0] += B1;

  float z = XP[0];
  float gate = 0.5f * z * (1.0f + erff(z * 0.70710678118654752440f)); // exact GELU
  float XG[16];
  #pragma unroll
  for (int d=0; d<16; ++d) XG[d] = gate * XP[d];

  #pragma unroll
  for (int k=0;k<16;++k){ GPV[k]=0.0f; JNV[k]=0.0f; }
  #pragma unroll
  for (int i=0;i<16;++i){
    #pragma unroll
    for (int j=0;j<16;++j){
      const float p = XN[i]*XG[j];
      { const int s = ga::GP.s[i][j]; if (s) GPV[ga::GP.k[i][j]] += (s>0)? p : -p; }
      { const int s = ga::JN.s[i][j]; if (s) JNV[ga::JN.k[i][j]] += (s>0)? p : -p; }
    }
  }
  #pragma unroll
  for (int k=0;k<16;++k) JNV[k] *= XN[15];
}

// ---------------------------------------------------------------------------
// Main kernel: full 256-token blocks; final fused linear pair + residual via
// chained V_WMMA_F32_16X16X4_F32 (wave32, EXEC all-ones guaranteed).
// ---------------------------------------------------------------------------
__global__ __launch_bounds__(BLOCK)
void mvffn_wmma_kernel(const float* __restrict__ x,
                       const float* __restrict__ w1, const float* __restrict__ v1, const float* __restrict__ b1,
                       const float* __restrict__ wg, const float* __restrict__ vg, const float* __restrict__ bg,
                       const float* __restrict__ wj, const float* __restrict__ vj, const float* __restrict__ bj,
                       float* __restrict__ out){
  // stride 18 floats: keeps (row*18 + 4q + {0,2})*4 byte addrs 8B-aligned for paired reads
  __shared__ float lx [BLOCK*18];
  __shared__ float lgp[BLOCK*18];
  __shared__ float ljn[BLOCK*18];

  const int tid = threadIdx.x;
  const size_t blockTok = (size_t)blockIdx.x * BLOCK;
  const float* xb = x + blockTok*16;

  // coalesced b128 loads -> LDS transpose to token rows
  #pragma unroll
  for (int q=0;q<4;++q){
    int m = q*BLOCK + tid;               // float4 id within block (0..1023)
    float4 v = ((const float4*)xb)[m];
    int trow = m >> 2, c0 = (m & 3) << 2;
    float* p = &lx[trow*18 + c0];
    p[0]=v.x; p[1]=v.y; p[2]=v.z; p[3]=v.w;
  }

  float W1[5], V1[4], WG[5], VG[4], WJ[5], VJ[4];
  #pragma unroll
  for (int k=0;k<5;++k){ W1[k]=w1[k]; WG[k]=wg[k]; WJ[k]=wj[k]; }
  #pragma unroll
  for (int k=0;k<4;++k){ V1[k]=v1[k]; VG[k]=vg[k]; VJ[k]=vj[k]; }
  const float B1=b1[0], BG=bg[0], BJ=bj[0];

  __syncthreads();

  float X[16];
  #pragma unroll
  for (int c=0;c<16;++c) X[c] = lx[tid*18 + c];

  float GPV[16], JNV[16];
  token_core(X, W1, V1, B1, GPV, JNV);

  #pragma unroll
  for (int c=0;c<16;++c){ lgp[tid*18+c]=GPV[c]; ljn[tid*18+c]=JNV[c]; }
  __syncthreads();

  // ---- WMMA stage: each wave owns 32 tokens = two 16x16 tiles ----
  const int lane = tid & 31;
  const int col  = lane & 15;               // N index
  const int rb   = (lane < 16) ? 0 : 2;     // K sub-row offset for this half-wave
  const int waveBase = (tid >> 5) << 5;     // block-local token base of this wave

  // B fragments for Mg and Mj (K=16 split into 4 fragments of K=4); layout mirrors
  // A: vgpr0 = row 4q+rb, vgpr1 = row 4q+rb+1, col = lane&15
  v2f Bg[4], Bj[4];
  #pragma unroll
  for (int q=0;q<4;++q){
    Bg[q].x = mval(4*q+rb,   col, WG[0],WG[1],WG[2],WG[3],WG[4], VG[1],VG[2],VG[3]);
    Bg[q].y = mval(4*q+rb+1, col, WG[0],WG[1],WG[2],WG[3],WG[4], VG[1],VG[2],VG[3]);
    Bj[q].x = mval(4*q+rb,   col, WJ[0],WJ[1],WJ[2],WJ[3],WJ[4], VJ[1],VJ[2],VJ[3]);
    Bj[q].y = mval(4*q+rb+1, col, WJ[0],WJ[1],WJ[2],WJ[3],WJ[4], VJ[1],VJ[2],VJ[3]);
  }

  #pragma unroll
  for (int tile=0; tile<2; ++tile){
    const int tb = waveBase + tile*16;
    // A fragments: lanes 0-15 hold token=lane rows K={4q,4q+1}; lanes 16-31 K={4q+2,4q+3}
    v2f Ag[4], Aj[4];
    const int arow = tb + col;
    #pragma unroll
    for (int q=0;q<4;++q){
      const float* pg = &lgp[arow*18 + 4*q + rb];
      Ag[q].x = pg[0]; Ag[q].y = pg[1];
      const float* pj = &ljn[arow*18 + 4*q + rb];
      Aj[q].x = pj[0]; Aj[q].y = pj[1];
    }
    // C accumulator = residual x + (bg+bj) on component 0
    v8f acc;
    const int row0 = tb + ((lane < 16) ? 0 : 8);
    const float cb = (col == 0) ? (BG + BJ) : 0.0f;
    #pragma unroll
    for (int r=0;r<8;++r) acc[r] = lx[(row0 + r)*18 + col] + cb;

    #pragma unroll
    for (int q=0;q<4;++q)
      acc = __builtin_amdgcn_wmma_f32_16x16x4_f32(false, Ag[q], false, Bg[q],
                                                  (short)0, acc, false, false);
    #pragma unroll
    for (int q=0;q<4;++q)
      acc = __builtin_amdgcn_wmma_f32_16x16x4_f32(false, Aj[q], false, Bj[q],
                                                  (short)0, acc, false, false);
    // stores: each b32 store covers two contiguous 64B token rows -> coalesced
    #pragma unroll
    for (int r=0;r<8;++r)
      out[(blockTok + (size_t)(row0 + r))*16 + col] = acc[r];
  }
}

// ---------------------------------------------------------------------------
// Scalar tail kernel (no WMMA -> safe with partial EXEC)
// ---------------------------------------------------------------------------
__device__ __forceinline__ void mv_acc(float O[16], const float S[16],
                                       const float W[5], const float V[4], float b){
  #pragma unroll
  for (int d=0;d<16;++d){
    float t = W[ga::GRADE[d]] * S[d];
    if (ga::PRE[d] >= 0 && ga::GRADE[d] <= 3) t += V[ga::GRADE[d]] * S[ga::PRE[d]];
    O[d] += t;
  }
  O[0] += b;
}

__global__ __launch_bounds__(BLOCK)
void mvffn_tail_kernel(const float* __restrict__ x,
                       const float* __restrict__ w1, const float* __restrict__ v1, const float* __restrict__ b1,
                       const float* __restrict__ wg, const float* __restrict__ vg, const float* __restrict__ bg,
                       const float* __restrict__ wj, const float* __restrict__ vj, const float* __restrict__ bj,
                       float* __restrict__ out, long long start, long long ntok){
  long long t = start + (long long)blockIdx.x*BLOCK + threadIdx.x;
  if (t >= ntok) return;
  float W1[5], V1[4], WG[5], VG[4], WJ[5], VJ[4];
  #pragma unroll
  for (int k=0;k<5;++k){ W1[k]=w1[k]; WG[k]=wg[k]; WJ[k]=wj[k]; }
  #pragma unroll
  for (int k=0;k<4;++k){ V1[k]=v1[k]; VG[k]=vg[k]; VJ[k]=vj[k]; }
  float X[16];
  const float* px = x + (size_t)t*16;
  #pragma unroll
  for (int c=0;c<16;++c) X[c] = px[c];
  float GPV[16], JNV[16];
  token_core(X, W1, V1, b1[0], GPV, JNV);
  float O[16];
  #pragma unroll
  for (int d=0;d<16;++d) O[d] = X[d];
  mv_acc(O, GPV, WG, VG, bg[0]);
  mv_acc(O, JNV, WJ, VJ, bj[0]);
  float* po = out + (size_t)t*16;
  #pragma unroll
  for (int d=0;d<16;++d) po[d] = O[d];
}

extern "C" void kernel_launch(void* const* d_in, const int* in_sizes, int n_in,
                              void* d_out, int out_size, void* d_ws, size_t ws_size,
                              hipStream_t stream){
  const float* x  = (const float*)d_in[0];
  const float* w1 = (const float*)d_in[1];
  const float* v1 = (const float*)d_in[2];
  const float* b1 = (const float*)d_in[3];
  const float* wg = (const float*)d_in[4];
  const float* vg = (const float*)d_in[5];
  const float* bg = (const float*)d_in[6];
  const float* wj = (const float*)d_in[7];
  const float* vj = (const float*)d_in[8];
  const float* bj = (const float*)d_in[9];
  float* out = (float*)d_out;

  long long ntok  = (long long)in_sizes[0] / 16;
  long long nfull = ntok / BLOCK;                  // 1024 for the reference shape
  if (nfull > 0)
    mvffn_wmma_kernel<<<(unsigned)nfull, BLOCK, 0, stream>>>(
        x, w1, v1, b1, wg, vg, bg, wj, vj, bj, out);
  long long tail = ntok - nfull*BLOCK;
  if (tail > 0)
    mvffn_tail_kernel<<<(unsigned)((tail + BLOCK - 1)/BLOCK), BLOCK, 0, stream>>>(
        x, w1, v1, b1, wg, vg, bg, wj, vj, bj, out, nfull*BLOCK, ntok);
}